// CenterLoss_55198919688422
// MI455X (gfx1250) — compile-verified
//
#include <hip/hip_runtime.h>

// CDNA5 / gfx1250, wave32. Memory-bound center-loss: one pass over x (64MiB),
// dots vs ALL 19 centers on the matrix pipe (V_WMMA_F32_16X16X4_F32, exact f32),
// ssq on VALU, label-select afterwards.

typedef float v2f __attribute__((ext_vector_type(2)));
typedef float v8f __attribute__((ext_vector_type(8)));

#define C_DIM   256
#define HW      16384          // 128*128
#define NPIX    65536          // 4*128*128
#define NCLS    19
#define EPSV    1e-12f
#define NCHUNK  64             // 256 channels / K=4
#define BTAB    (NCHUNK * 2 * 32)   // 4096 float2 entries (two 16-class tiles)

// d_ws layout (floats):
//   [0,    8192) : B-operand table (4096 float2, WMMA-register-swizzled)
//   [8192, 8224) : cnsq[32] (||cn_k||^2, zero-padded)
//   [8224, 8480) : per-block partial sums (256)

__global__ void prep_centers(const float* __restrict__ centers,
                             float2* __restrict__ wsB,
                             float* __restrict__ wsCnsq) {
    __shared__ float sInv[NCLS];
    int tid = threadIdx.x;
    if (tid < NCLS) {
        float s = 0.f;
        const float* row = centers + tid * C_DIM;
        for (int c = 0; c < C_DIM; ++c) { float v = row[c]; s = fmaf(v, v, s); }
        float denom = fmaxf(sqrtf(s), EPSV);
        float inv = 1.f / denom;
        sInv[tid] = inv;
        wsCnsq[tid] = s * inv * inv;          // ||centers_k/denom||^2
    }
    if (tid >= NCLS && tid < 32) wsCnsq[tid] = 0.f;
    __syncthreads();
    // B-operand layout for V_WMMA_F32_16X16X4_F32 (mirror of documented A 16x4):
    //   lane l<16 : col N=l,    K={0,1};  lane l>=16 : col N=l-16, K={2,3}
    for (int idx = tid; idx < BTAB; idx += blockDim.x) {
        int ch = idx >> 6;         // K-chunk (4 channels each)
        int r  = idx & 63;
        int t  = r >> 5;           // class tile 0: classes 0-15, tile 1: 16-31
        int l  = r & 31;           // lane
        int k  = (l & 15) + (t << 4);
        int c  = (ch << 2) + ((l >> 4) << 1);
        float2 v = make_float2(0.f, 0.f);
        if (k < NCLS) {
            float inv = sInv[k];
            v.x = centers[k * C_DIM + c]     * inv;
            v.y = centers[k * C_DIM + c + 1] * inv;
        }
        wsB[idx] = v;
    }
}

__global__ __launch_bounds__(256) void center_loss_main(
        const float* __restrict__ x,
        const int*  __restrict__ labels,
        const float2* __restrict__ wsB,
        const float* __restrict__ wsCnsq,
        float* __restrict__ partials) {
    __shared__ float2 sB[BTAB];     // 32 KB
    __shared__ float  sCnsq[32];
    __shared__ float  sRed[8];

    int tid = threadIdx.x;
    {   // stage B table into LDS (float4 traffic)
        const float4* s4 = (const float4*)wsB;
        float4* d4 = (float4*)sB;
        for (int i = tid; i < BTAB / 2; i += 256) d4[i] = s4[i];
        if (tid < 32) sCnsq[tid] = wsCnsq[tid];
    }
    __syncthreads();

    int lane = tid & 31;
    int wv   = tid >> 5;
    int m    = lane & 15;     // pixel row within 16-pixel tile
    int hi   = lane >> 4;     // K-half: channels {0,1} vs {2,3} of each chunk
    int gw   = blockIdx.x * 8 + wv;
    int totalWaves = gridDim.x * 8;

    float distAcc = 0.f;

    for (int tile = gw; tile < NPIX / 16; tile += totalWaves) {
        int p0 = tile * 16;
        int n  = p0 >> 14;                 // image index (HW = 2^14)
        int q  = (p0 & (HW - 1)) + m;      // pixel within image (tiles never cross images)
        const float* ptr = x + (size_t)n * C_DIM * HW + (size_t)(2 * hi) * HW + q;
        int k = labels[p0 + m];

        v8f acc0 = {};                      // dots vs classes 0-15
        v8f acc1 = {};                      // dots vs classes 16-31 (19-31 zero)
        float ssq = 0.f;

        #pragma unroll 4
        for (int ch = 0; ch < NCHUNK; ++ch) {
            float a0 = ptr[0];              // channel 4*ch + 2*hi
            float a1 = ptr[HW];             // channel 4*ch + 2*hi + 1
            if ((ch & 3) == 0 && ch + 8 < NCHUNK)
                __builtin_prefetch(ptr + (size_t)32 * HW, 0, 0);  // global_prefetch
            float2 b0 = sB[ch * 64 + lane];
            float2 b1 = sB[ch * 64 + 32 + lane];
            v2f a   = {a0, a1};
            v2f vb0 = {b0.x, b0.y};
            v2f vb1 = {b1.x, b1.y};
            acc0 = __builtin_amdgcn_wmma_f32_16x16x4_f32(
                       false, a, false, vb0, (short)0, acc0, false, false);
            acc1 = __builtin_amdgcn_wmma_f32_16x16x4_f32(
                       false, a, false, vb1, (short)0, acc1, false, false);
            ssq = fmaf(a0, a0, ssq);
            ssq = fmaf(a1, a1, ssq);
            ptr += (size_t)4 * HW;
        }

        // ssq for pixel m is split across lanes m and m+16
        ssq += __shfl_xor(ssq, 16, 32);

        // extract dot = D[m][k]: D row m lives in acc VGPR (m&7), lane (n + 16*(m>>3))
        int ktile   = k >> 4;
        int srcLane = (k & 15) + ((m >> 3) << 4);
        int jm      = m & 7;
        float dotv = 0.f;
        #pragma unroll
        for (int j = 0; j < 8; ++j) {
            float t0 = __shfl(acc0[j], srcLane, 32);
            float t1 = __shfl(acc1[j], srcLane, 32);
            float sel = (ktile == 0) ? t0 : t1;
            dotv = (j == jm) ? sel : dotv;
        }

        float denom = fmaxf(sqrtf(ssq), EPSV);
        float inv   = 1.f / denom;
        float dist  = ssq * inv * inv + sCnsq[k] - 2.f * dotv * inv;
        distAcc += (lane < 16) ? dist : 0.f;   // lanes 16-31 are duplicates
    }

    // wave reduction
    #pragma unroll
    for (int off = 16; off >= 1; off >>= 1)
        distAcc += __shfl_xor(distAcc, off, 32);
    if (lane == 0) sRed[wv] = distAcc;
    __syncthreads();
    if (tid == 0) {
        float s = 0.f;
        #pragma unroll
        for (int i = 0; i < 8; ++i) s += sRed[i];
        partials[blockIdx.x] = s;
    }
}

__global__ void final_reduce(const float* __restrict__ partials,
                             float* __restrict__ out) {
    __shared__ float sRed[8];
    int tid = threadIdx.x;
    float v = partials[tid];               // 256 partials, 256 threads
    #pragma unroll
    for (int off = 16; off >= 1; off >>= 1)
        v += __shfl_xor(v, off, 32);
    if ((tid & 31) == 0) sRed[tid >> 5] = v;
    __syncthreads();
    if (tid == 0) {
        float s = 0.f;
        #pragma unroll
        for (int i = 0; i < 8; ++i) s += sRed[i];
        out[0] = s * (1.0f / 65536.0f);    // mean over all pixels
    }
}

extern "C" void kernel_launch(void* const* d_in, const int* in_sizes, int n_in,
                              void* d_out, int out_size, void* d_ws, size_t ws_size,
                              hipStream_t stream) {
    const float* x       = (const float*)d_in[0];   // [4,256,128,128] f32
    const float* centers = (const float*)d_in[1];   // [19,256] f32
    const int*   labels  = (const int*)d_in[2];     // [4,128,128] int

    float*  ws     = (float*)d_ws;
    float2* wsB    = (float2*)ws;          // 4096 float2 = 32 KB
    float*  wsCnsq = ws + 8192;            // 32 floats
    float*  wsPart = ws + 8224;            // 256 floats

    prep_centers<<<1, 256, 0, stream>>>(centers, wsB, wsCnsq);
    center_loss_main<<<256, 256, 0, stream>>>(x, labels, wsB, wsCnsq, wsPart);
    final_reduce<<<1, 256, 0, stream>>>(wsPart, (float*)d_out);
}